// MS_PartGuidedBlock_16896401343295
// MI455X (gfx1250) — compile-verified
//
#include <hip/hip_runtime.h>
#include <hip/hip_bf16.h>

typedef __attribute__((ext_vector_type(2))) float v2f;
typedef __attribute__((ext_vector_type(8))) float v8f;

// Problem constants (from reference setup): B=8, C=256, H=64, W=48, N=3072, 8 classes.
#define NB 8
#define NC 256
#define NPIX 3072
#define NCLS 8

// ---------------------------------------------------------------------------
// Kernel 1: per-batch label histogram -> cnt[b][k]
// ---------------------------------------------------------------------------
__global__ void count_kernel(const int* __restrict__ lab, int* __restrict__ cnt) {
    __shared__ int hist[NCLS];
    const int b = blockIdx.x;
    const int tid = threadIdx.x;
    if (tid < NCLS) hist[tid] = 0;
    __syncthreads();
    for (int n = tid; n < NPIX; n += 256) {
        atomicAdd(&hist[lab[b * NPIX + n]], 1);
    }
    __syncthreads();
    if (tid < NCLS) cnt[b * NCLS + tid] = hist[tid];
}

// ---------------------------------------------------------------------------
// Kernel 2: class sums via V_WMMA_F32_16X16X4_F32.
//   S[b,k,c] = sum_{m : lab[b,m]==k} x[b,c,m]
// One block per (b, 16-channel tile); 4 waves split the K=3072 pixel axis.
// A = one-hot(class x pixel) fp32 (exact), B = x chunk (pixel x channel).
// ISA layouts (16x16x4 f32):
//   A: lane L holds K = 2*(L/16), 2*(L/16)+1 for row M = L%16   (2 VGPRs)
//   B: lane L holds K = 2*(L/16), 2*(L/16)+1 for col N = L%16   (2 VGPRs)
//   C/D: VGPR j, lane L -> (M = j + 8*(L/16), N = L%16)
// Classes 8..15 have all-zero A rows -> lanes 16..31 accumulate exact zeros.
// ---------------------------------------------------------------------------
__global__ void class_sum_wmma(const float* __restrict__ x,
                               const int* __restrict__ lab,
                               float* __restrict__ S) {
    __shared__ float red[4][NCLS][16];

    const int blk  = blockIdx.x;       // 0..127
    const int b    = blk >> 4;
    const int tile = blk & 15;
    const int c0   = tile * 16;

    const int tid  = threadIdx.x;      // 128 threads = 4 waves
    const int wave = tid >> 5;
    const int lane = tid & 31;
    const int half = lane >> 4;        // 0: K pair {0,1}, 1: K pair {2,3}
    const int lc   = lane & 15;        // M (class) for A, N (channel) for B

    const float* xrow = x + (size_t)(b * NC + c0 + lc) * NPIX;
    const int*   lrow = lab + b * NPIX;

    v8f acc = {};
    const int m0 = wave * (NPIX / 4);  // 768-pixel slice per wave
    for (int m = m0; m < m0 + NPIX / 4; m += 4) {
        const int2   lb = *(const int2*)(lrow + m + 2 * half);
        const float2 xv = *(const float2*)(xrow + m + 2 * half);
        v2f a, bv;
        a.x  = (lb.x == lc) ? 1.0f : 0.0f;
        a.y  = (lb.y == lc) ? 1.0f : 0.0f;
        bv.x = xv.x;
        bv.y = xv.y;
        // 8 args: (neg_a, A, neg_b, B, c_mod, C, reuse_a, reuse_b)
        acc = __builtin_amdgcn_wmma_f32_16x16x4_f32(
            false, a, false, bv, (short)0, acc, false, false);
    }

    if (half == 0) {                   // lanes 0..15 hold classes 0..7
#pragma unroll
        for (int j = 0; j < NCLS; ++j) red[wave][j][lc] = acc[j];
    }
    __syncthreads();

    if (tid < NCLS * 16) {
        const int k = tid >> 4;
        const int c = tid & 15;
        const float s = red[0][k][c] + red[1][k][c] + red[2][k][c] + red[3][k][c];
        S[(b * NCLS + k) * NC + c0 + c] = s;
    }
}

// ---------------------------------------------------------------------------
// Kernel 3: lookup table T[b,k,c]:
//   k==0 : global mean  (all-ones similarity row, denom = N)
//   k>=1 : class mean   (denom = max(count, EPS) to match reference)
// ---------------------------------------------------------------------------
__global__ void table_kernel(const float* __restrict__ S,
                             const int* __restrict__ cnt,
                             float* __restrict__ T) {
    const int b = blockIdx.x;
    const int c = threadIdx.x;         // 256 channels
    float s[NCLS];
    float tot = 0.0f;
#pragma unroll
    for (int k = 0; k < NCLS; ++k) {
        s[k] = S[(b * NCLS + k) * NC + c];
        tot += s[k];
    }
    T[(b * NCLS + 0) * NC + c] = tot * (1.0f / (float)NPIX);
#pragma unroll
    for (int k = 1; k < NCLS; ++k) {
        const float d = fmaxf((float)cnt[b * NCLS + k], 1e-12f);
        T[(b * NCLS + k) * NC + c] = s[k] / d;
    }
}

// ---------------------------------------------------------------------------
// Kernel 4: streaming scatter  out[b,c,n] = T[b, lab[b,n], c]
// grid = (n-tiles=12, b=8, c-quads=4); 256 threads cover 256 consecutive n,
// so every store of the c-loop is a contiguous 1KB line. T slice in LDS.
// ---------------------------------------------------------------------------
__global__ void scatter_kernel(const int* __restrict__ lab,
                               const float* __restrict__ T,
                               float* __restrict__ out) {
    __shared__ float tt[NCLS * 64];    // 8 classes x 64 channels of this quad
    const int tid = threadIdx.x;
    const int b   = blockIdx.y;
    const int n   = blockIdx.x * 256 + tid;
    const int c0  = blockIdx.z * 64;

    for (int i = tid; i < NCLS * 64; i += 256) {
        const int k = i >> 6;
        const int c = i & 63;
        tt[i] = T[(b * NCLS + k) * NC + c0 + c];
    }
    const int L = lab[b * NPIX + n];
    __syncthreads();

    const float* trow  = &tt[L * 64];
    float*       obase = out + (size_t)(b * NC + c0) * NPIX + n;
#pragma unroll 4
    for (int c = 0; c < 64; ++c) {
        obase[(size_t)c * NPIX] = trow[c];
    }
}

// ---------------------------------------------------------------------------
extern "C" void kernel_launch(void* const* d_in, const int* in_sizes, int n_in,
                              void* d_out, int out_size, void* d_ws, size_t ws_size,
                              hipStream_t stream) {
    const float* x   = (const float*)d_in[0];   // (B, C, H, W) fp32
    const int*   lab = (const int*)d_in[1];     // (B, H, W) int32
    float*       out = (float*)d_out;           // (B, C, H, W) fp32

    // Workspace layout: S (B*8*256 f32) | cnt (B*8 i32) | T (B*8*256 f32)
    float* S   = (float*)d_ws;
    int*   cnt = (int*)(S + NB * NCLS * NC);
    float* T   = (float*)(cnt + NB * NCLS);

    count_kernel<<<NB, 256, 0, stream>>>(lab, cnt);
    class_sum_wmma<<<NB * (NC / 16), 128, 0, stream>>>(x, lab, S);
    table_kernel<<<NB, NC, 0, stream>>>(S, cnt, T);
    scatter_kernel<<<dim3(NPIX / 256, NB, 4), 256, 0, stream>>>(lab, T, out);
}